// EncoderClassifierMamba_23527830847742
// MI455X (gfx1250) — compile-verified
//
#include <hip/hip_runtime.h>

#define B_      128
#define S_      37
#define T_      2048
#define D_      256
#define ST_     8
#define C_      2
#define F_      74         // 2*S (measurements + mask features)
#define MERGED_ 264        // D + ST
#define CH_     16         // time chunks
#define TT_     128        // timesteps per chunk (T_ / CH_)
#define NF_     76         // 74 feature sums + masked-time-sum + denom
#define KP_     76         // padded K for pooled WMMA (74 real + 2 zeros)

typedef __attribute__((ext_vector_type(2))) float v2f;
typedef __attribute__((ext_vector_type(8))) float v8f;

__device__ __forceinline__ v8f wmma_f32_k4(v2f a, v2f b, v8f c) {
  return __builtin_amdgcn_wmma_f32_16x16x4_f32(
      /*neg_a=*/false, a, /*neg_b=*/false, b,
      /*c_mod=*/(short)0, c, /*reuse_a=*/false, /*reuse_b=*/false);
}

// ---------------- Stage 1: masked partial reductions over time ----------------
// One block = (batch b, time-chunk ch). Streams x and sensor_mask exactly once
// through LDS; deterministic per-chunk partial sums (no atomics).
__global__ void reduce_kernel(const float* __restrict__ x,
                              const int*   __restrict__ sm,
                              const float* __restrict__ timev,
                              float* __restrict__ part) {
  const int b    = blockIdx.x;
  const int ch   = blockIdx.y;
  const int t0   = ch * TT_;
  const int tid  = threadIdx.x;        // 0..127 (4 waves of 32)
  const int wave = tid >> 5;
  const int lane = tid & 31;

  __shared__ float xs[S_ * TT_];       // 18.5 KB
  __shared__ float ms[S_ * TT_];       // 18.5 KB
  __shared__ float tvals[TT_];
  __shared__ float validf[TT_];

  // Coalesced tile load: contiguous in t for each sensor row.
  #pragma unroll 1
  for (int s = 0; s < S_; ++s) {
    const int g = (b * S_ + s) * T_ + t0 + tid;
    if (s + 1 < S_) {
      __builtin_prefetch(&x[g + T_], 0, 3);    // global_prefetch_b8 (next row)
      __builtin_prefetch(&sm[g + T_], 0, 3);
    }
    xs[s * TT_ + tid] = x[g];
    ms[s * TT_ + tid] = (float)sm[g];
  }
  tvals[tid] = timev[b * T_ + t0 + tid];
  __syncthreads();

  // Timestep validity: any nonzero feature among the 74.
  int v = 0;
  #pragma unroll 1
  for (int s = 0; s < S_; ++s) {
    v |= (xs[s * TT_ + tid] != 0.0f);
    v |= (ms[s * TT_ + tid] != 0.0f);
  }
  validf[tid] = (float)v;
  __syncthreads();

  // Per-feature masked sums; f==74 -> masked time sum, f==75 -> denom.
  float* dst = part + (b * CH_ + ch) * NF_;
  for (int f = wave; f < NF_; f += 4) {
    float p = 0.0f;
    #pragma unroll
    for (int j = 0; j < TT_ / 32; ++j) {
      const int t = lane + 32 * j;
      float val;
      if (f < S_)        val = xs[f * TT_ + t];
      else if (f < F_)   val = ms[(f - S_) * TT_ + t];
      else if (f == F_)  val = tvals[t];
      else               val = 1.0f;
      p += validf[t] * val;
    }
    p += __shfl_down(p, 16);
    p += __shfl_down(p, 8);
    p += __shfl_down(p, 4);
    p += __shfl_down(p, 2);
    p += __shfl_down(p, 1);
    if (lane == 0) dst[f] = p;
  }
}

// ------- Stage 2a: fold chunk partials; emit padded A-matrix + static head -------
// Precomputes 1/denom once per batch so the WMMA epilogue is a pure multiply
// (avoids 8 IEEE division expansions per lane in the hot epilogue).
__global__ void head_sum_kernel(const float* __restrict__ part,
                                const float* __restrict__ statics,
                                const float* __restrict__ W_static,
                                const float* __restrict__ b_static,
                                float* __restrict__ xsp,      // [B][KP_], zero padded
                                float* __restrict__ invdv,    // [B] 1/denom
                                float* __restrict__ tmeanv,   // [B] tsum/denom
                                float* __restrict__ comb) {   // writes cols 256..263
  const int b   = blockIdx.x;
  const int tid = threadIdx.x;         // 128 threads
  __shared__ float sfeat[NF_];

  if (tid < NF_) {
    float a = 0.0f;
    #pragma unroll 1
    for (int c = 0; c < CH_; ++c) a += part[(b * CH_ + c) * NF_ + tid];
    sfeat[tid] = a;
  }
  __syncthreads();

  if (tid < KP_) xsp[b * KP_ + tid] = (tid < F_) ? sfeat[tid] : 0.0f;
  if (tid == 0) {
    const float denom = fmaxf(sfeat[F_ + 1], 1e-9f);
    invdv[b]  = 1.0f / denom;
    tmeanv[b] = sfeat[F_] / denom;
  }
  if (tid < ST_) {
    float acc = b_static[tid];
    #pragma unroll
    for (int i = 0; i < ST_; ++i)
      acc += statics[b * ST_ + i] * W_static[i * ST_ + tid];
    comb[b * MERGED_ + D_ + tid] = acc;
  }
}

// ------- Stage 2b: pooled = (Xs @ W_sensor)/denom + biases, via fp32 WMMA -------
// One wave per 16x16 tile of [128 x 256]. K padded to 76 (A holds exact zeros
// at k=74,75; W row index clamped so 0 * finite == 0).
__global__ void pooled_wmma_kernel(const float* __restrict__ xsp,
                                   const float* __restrict__ W_sensor,
                                   const float* __restrict__ b_sensor,
                                   const float* __restrict__ b_time,
                                   const float* __restrict__ W_time,
                                   const float* __restrict__ invdv,
                                   const float* __restrict__ tmeanv,
                                   float* __restrict__ comb) {
  const int n0   = blockIdx.x * 16;    // 16 N tiles (256 cols)
  const int m0   = blockIdx.y * 16;    // 8 M tiles (128 rows)
  const int lane = threadIdx.x;
  const int ml   = lane & 15;
  const int hi   = lane >> 4;
  const int n    = n0 + ml;

  const float* arow = xsp + (m0 + ml) * KP_;
  v8f acc = {0.f, 0.f, 0.f, 0.f, 0.f, 0.f, 0.f, 0.f};

  #pragma unroll 1
  for (int k0 = 0; k0 < KP_; k0 += 4) {        // 19 steps
    const int kb  = k0 + hi * 2;
    const int k0c = (kb     < F_) ? kb     : (F_ - 1);
    const int k1c = (kb + 1 < F_) ? kb + 1 : (F_ - 1);
    v2f a, bv;
    a.x  = arow[kb];
    a.y  = arow[kb + 1];
    bv.x = W_sensor[k0c * D_ + n];
    bv.y = W_sensor[k1c * D_ + n];
    acc = wmma_f32_k4(a, bv, acc);
  }

  const float bs = b_sensor[n], bt = b_time[n], wt = W_time[n];
  #pragma unroll
  for (int r = 0; r < 8; ++r) {
    const int m = m0 + r + hi * 8;
    comb[m * MERGED_ + n] = acc[r] * invdv[m] + bs + bt + tmeanv[m] * wt;
  }
}

// ---------------- Stage 3: ReLU(comb @ W_merge + b) via fp32 WMMA ----------------
// Column index clamped (no EXEC divergence in the K loop); partial unroll
// clusters loads and issues WMMAs back-to-back.
__global__ void merge_wmma_kernel(const float* __restrict__ comb,
                                  const float* __restrict__ W_merge,
                                  const float* __restrict__ b_merge,
                                  float* __restrict__ comb2) {
  const int n0   = blockIdx.x * 16;    // 17 N tiles (264 cols, last partial)
  const int m0   = blockIdx.y * 16;    // 8 M tiles
  const int lane = threadIdx.x;
  const int ml   = lane & 15;
  const int hi   = lane >> 4;
  const int n    = n0 + ml;
  const int ncl  = (n < MERGED_) ? n : (MERGED_ - 1);   // clamp, not predicate

  const float* arow = comb + (m0 + ml) * MERGED_;
  v8f acc = {0.f, 0.f, 0.f, 0.f, 0.f, 0.f, 0.f, 0.f};

  #pragma unroll 1
  for (int k0 = 0; k0 < MERGED_; k0 += 8) {    // 33 trips, 2 WMMA each
    const int kb = k0 + hi * 2;
    v2f a0, b0, a1, b1;
    a0.x = arow[kb];
    a0.y = arow[kb + 1];
    b0.x = W_merge[kb * MERGED_ + ncl];
    b0.y = W_merge[(kb + 1) * MERGED_ + ncl];
    a1.x = arow[kb + 4];
    a1.y = arow[kb + 5];
    b1.x = W_merge[(kb + 4) * MERGED_ + ncl];
    b1.y = W_merge[(kb + 5) * MERGED_ + ncl];
    acc = wmma_f32_k4(a0, b0, acc);
    acc = wmma_f32_k4(a1, b1, acc);
  }

  if (n < MERGED_) {
    const float bias = b_merge[n];
    #pragma unroll
    for (int r = 0; r < 8; ++r) {
      const int m = m0 + r + hi * 8;
      comb2[m * MERGED_ + n] = fmaxf(acc[r] + bias, 0.0f);
    }
  }
}

// ---------------- Stage 4: classifier [128,264] @ [264,2] ----------------
__global__ void cls_kernel(const float* __restrict__ comb2,
                           const float* __restrict__ W_cls,
                           const float* __restrict__ b_cls,
                           float* __restrict__ out) {
  const int tid = blockIdx.x * blockDim.x + threadIdx.x;
  if (tid >= B_ * C_) return;
  const int b = tid >> 1, c = tid & 1;
  float acc = b_cls[c];
  #pragma unroll 1
  for (int k = 0; k < MERGED_; ++k)
    acc += comb2[b * MERGED_ + k] * W_cls[k * C_ + c];
  out[b * C_ + c] = acc;
}

extern "C" void kernel_launch(void* const* d_in, const int* in_sizes, int n_in,
                              void* d_out, int out_size, void* d_ws, size_t ws_size,
                              hipStream_t stream) {
  const float* x        = (const float*)d_in[0];   // [B,S,T]
  const float* statics  = (const float*)d_in[1];   // [B,ST]
  const float* timev    = (const float*)d_in[2];   // [B,T]
  const int*   smask    = (const int*)  d_in[3];   // [B,S,T]
  const float* W_sensor = (const float*)d_in[4];   // [74,256]
  const float* b_sensor = (const float*)d_in[5];   // [256]
  const float* W_time   = (const float*)d_in[6];   // [1,256]
  const float* b_time   = (const float*)d_in[7];   // [256]
  const float* W_static = (const float*)d_in[8];   // [8,8]
  const float* b_static = (const float*)d_in[9];   // [8]
  const float* W_merge  = (const float*)d_in[10];  // [264,264]
  const float* b_merge  = (const float*)d_in[11];  // [264]
  const float* W_cls    = (const float*)d_in[12];  // [264,2]
  const float* b_cls    = (const float*)d_in[13];  // [2]
  float* out = (float*)d_out;

  float* ws     = (float*)d_ws;
  float* part   = ws;                                 // B*CH*NF  = 155,648 f
  float* xsp    = part   + (size_t)B_ * CH_ * NF_;    // B*KP     =   9,728 f
  float* invdv  = xsp    + (size_t)B_ * KP_;          // B        =     128 f
  float* tmeanv = invdv  + B_;                        // B        =     128 f
  float* comb   = tmeanv + B_;                        // B*MERGED =  33,792 f
  float* comb2  = comb   + (size_t)B_ * MERGED_;      // B*MERGED =  33,792 f

  reduce_kernel<<<dim3(B_, CH_), 128, 0, stream>>>(x, smask, timev, part);
  head_sum_kernel<<<B_, 128, 0, stream>>>(part, statics, W_static, b_static,
                                          xsp, invdv, tmeanv, comb);
  pooled_wmma_kernel<<<dim3(D_ / 16, B_ / 16), 32, 0, stream>>>(
      xsp, W_sensor, b_sensor, b_time, W_time, invdv, tmeanv, comb);
  merge_wmma_kernel<<<dim3((MERGED_ + 15) / 16, B_ / 16), 32, 0, stream>>>(
      comb, W_merge, b_merge, comb2);
  cls_kernel<<<1, 256, 0, stream>>>(comb2, W_cls, b_cls, out);
}